// SeparableModulatedConv2d_65618510349005
// MI455X (gfx1250) — compile-verified
//
#include <hip/hip_runtime.h>

// Spatially-varying separable DoG on MI455X (gfx1250).
// Band-matmul formulation: per 16-pixel output tile,
//   out[c, j] = sum_t A[c,t] * B[t,j],  A = reflected input row segment (16x80),
//   B[t,j] = G_j[t-j] (banded per-pixel Gaussian weights, zero off-band),
// executed as 20 chained V_WMMA_F32_16X16X4_F32 per sigma (fp32-exact path).
// Staging uses CDNA5 async global->LDS copies (ASYNCcnt); intermediates are
// stored interleaved (H1,H2) so scattered transposed writes are b64-wide.

#define R_   31
#define KLEN 63
#define B_   4
#define C_   32
#define H_   512
#define W_   512
#define TS   16      // tile: 16 channels x 16 pixels
#define KTOT 80      // band columns padded to multiple of 4
#define ASTR 84      // A LDS row stride (floats): banks 20*c+t mod 64, conflict-free
#define WSTR 96      // weight-column stride: banks 31*j+t mod 64, conflict-free
#define WLEN 96      // 16-entry zero apron + 63 taps + tail zeros

typedef float v2f __attribute__((ext_vector_type(2)));
typedef float v8f __attribute__((ext_vector_type(8)));

__device__ __forceinline__ int reflect_idx(int i, int n) {
  // jnp.pad 'reflect': -1 -> 1, n -> n-2 (R_ < n so one fold suffices)
  if (i < 0) i = -i;
  if (i >= n) i = 2 * n - 2 - i;
  return i;
}

// CDNA5 async copy: global (per-lane 64-bit addr) -> LDS (per-lane 32-bit offset).
__device__ __forceinline__ void async_g2l_b32(const float* lds_dst, const float* gsrc) {
  unsigned lds_off = (unsigned)(uintptr_t)lds_dst;            // addr[31:0] = LDS offset
  asm volatile("global_load_async_to_lds_b32 %0, %1, off"
               :: "v"(lds_off), "v"((unsigned long long)(uintptr_t)gsrc)
               : "memory");
}
__device__ __forceinline__ void wait_async0() {
  asm volatile("s_wait_asynccnt 0x0" ::: "memory");
}

// Threads 0..31 each build one normalized truncated-Gaussian column in LDS:
// col = tid; pixel j = tid&15; sigma set s = tid>>4 (s==1 -> sigma*1.6).
// Column layout: [0..15]=0 apron, [16+k]=G[k] k=0..62, [79..95]=0.
__device__ __forceinline__ void gen_weights(float* Wlds, const float* __restrict__ sig,
                                            int sigBase, int sigStride, int tid) {
  if (tid < 32) {
    int j = tid & 15;
    float sg = sig[sigBase + j * sigStride];
    if (tid >= 16) sg *= 1.6f;                 // SIGMA_RATIO
    float rr = -0.5f / (sg * sg);
    float* col = Wlds + tid * WSTR;
    for (int i = 0; i < 16; ++i) col[i] = 0.0f;
    for (int i = 16 + KLEN; i < WLEN; ++i) col[i] = 0.0f;
    float sum = 0.0f;
    for (int k = 0; k < KLEN; ++k) {
      float r = (float)(k - R_);
      float e = __expf(rr * r * r);            // v_exp_f32 path
      e = (e >= 1e-3f) ? e : 0.0f;             // hard truncation (gauss_epsilon)
      sum += e;
      col[16 + k] = e;
    }
    float nrm = 1.0f / (sum + 1e-8f);
    for (int k = 0; k < KLEN; ++k) col[16 + k] *= nrm;
  }
}

// One banded 16x16x80 matmul: A (16 ch x 80 band cols) times banded weights.
// A-frag layout (16x4 f32): lane L holds M=L&15, K = 4*kk + 2*(L>>4) + {0,1}.
// B-frag layout (4x16 f32): lane L holds N=L&15, K = 4*kk + 2*(L>>4) + {0,1}.
__device__ __forceinline__ v8f band_mma_one(const float* __restrict__ A,
                                            const float* __restrict__ col,
                                            int lane, v8f acc) {
  int half = lane >> 4, jm = lane & 15;
  const float* arow = A + jm * ASTR;
#pragma unroll
  for (int kk = 0; kk < KTOT / 4; ++kk) {
    int t = kk * 4 + 2 * half;
    v2f a;  a.x = arow[t];        a.y = arow[t + 1];        // ds b64 (aligned)
    int wi = t - jm + 16;                                   // band index + apron
    v2f bb; bb.x = col[wi];       bb.y = col[wi + 1];
    acc = __builtin_amdgcn_wmma_f32_16x16x4_f32(
        /*neg_a=*/false, a, /*neg_b=*/false, bb,
        /*c_mod=*/(short)0, acc, /*reuse_a=*/false, /*reuse_b=*/false);
  }
  return acc;
}

// Pass 1: horizontal blur of x with G1 and G2; outputs stored TRANSPOSED and
// INTERLEAVED as [B, C, W, H, 2] so pass 2 reads pairs contiguously along H.
__global__ __launch_bounds__(64) void dog_pass1(
    const float* __restrict__ x, const float* __restrict__ sig,
    float* __restrict__ h12) {
  __shared__ float Alds[2][TS * ASTR];
  __shared__ float Wlds[32 * WSTR];

  int h  = blockIdx.x;
  int w0 = blockIdx.y * TS;
  int b  = blockIdx.z;
  int tid = threadIdx.x, wave = tid >> 5, lane = tid & 31;

  gen_weights(Wlds, sig, b * H_ * W_ + h * W_ + w0, 1, tid);

  // Stage this wave's 16-channel x-row band: async global -> LDS (coalesced).
  int c0 = wave * TS;
  const float* xrow = x + ((size_t)(b * C_ + c0) * H_ + h) * W_;
  float* A = Alds[wave];
#pragma unroll 4
  for (int i = 0; i < (TS * KTOT) / 32; ++i) {
    int idx = lane + 32 * i;
    int row = idx / KTOT, colp = idx % KTOT;
    int m = reflect_idx(w0 + colp - R_, W_);
    async_g2l_b32(A + row * ASTR + colp, xrow + (size_t)row * (H_ * W_) + m);
  }
  wait_async0();
  __syncthreads();

  int jm = lane & 15;
  v8f acc1 = {};
  v8f acc2 = {};
  acc1 = band_mma_one(A, Wlds + jm * WSTR,        lane, acc1);  // G1
  acc2 = band_mma_one(A, Wlds + (16 + jm) * WSTR, lane, acc2);  // G2

  int half = lane >> 4;
#pragma unroll
  for (int v = 0; v < 8; ++v) {
    int c = c0 + v + 8 * half;                                  // D layout: M=v+8*half
    size_t addr = ((size_t)(b * C_ + c) * W_ + (w0 + jm)) * H_ + h;  // transposed
    v2f pr; pr.x = acc1[v]; pr.y = acc2[v];
    *(v2f*)(h12 + 2 * addr) = pr;                               // one b64 per element pair
  }
}

// Pass 2: vertical blur (= horizontal over transposed interleaved layout);
// out = blur(h2, G2) - blur(h1, G1), written to [B,C,H,W].
__global__ __launch_bounds__(64) void dog_pass2(
    const float* __restrict__ h12, const float* __restrict__ sig,
    float* __restrict__ out) {
  __shared__ float Alds[2][2][TS * ASTR];
  __shared__ float Wlds[32 * WSTR];

  int w  = blockIdx.x;
  int h0 = blockIdx.y * TS;
  int b  = blockIdx.z;
  int tid = threadIdx.x, wave = tid >> 5, lane = tid & 31;

  gen_weights(Wlds, sig, b * H_ * W_ + h0 * W_ + w, W_, tid);

  int c0 = wave * TS;
  size_t rbase = ((size_t)(b * C_ + c0) * W_ + w) * (size_t)H_;
  float* A1 = Alds[wave][0];
  float* A2 = Alds[wave][1];
#pragma unroll 4
  for (int i = 0; i < (TS * KTOT) / 32; ++i) {
    int idx = lane + 32 * i;
    int row = idx / KTOT, colp = idx % KTOT;
    int m = reflect_idx(h0 + colp - R_, H_);
    size_t off2 = 2 * (rbase + (size_t)row * (W_ * H_) + m);   // contiguous in H
    async_g2l_b32(A1 + row * ASTR + colp, h12 + off2);         // H1 component
    async_g2l_b32(A2 + row * ASTR + colp, h12 + off2 + 1);     // H2 component
  }
  wait_async0();
  __syncthreads();

  int jm = lane & 15;
  v8f acc1 = {};
  v8f acc2 = {};
  acc1 = band_mma_one(A1, Wlds + jm * WSTR,        lane, acc1);  // G1 on h1
  acc2 = band_mma_one(A2, Wlds + (16 + jm) * WSTR, lane, acc2);  // G2 on h2

  int half = lane >> 4;
#pragma unroll
  for (int v = 0; v < 8; ++v) {
    int c = c0 + v + 8 * half;
    size_t addr = ((size_t)(b * C_ + c) * H_ + (h0 + jm)) * (size_t)W_ + w;
    out[addr] = acc2[v] - acc1[v];
  }
}

extern "C" void kernel_launch(void* const* d_in, const int* in_sizes, int n_in,
                              void* d_out, int out_size, void* d_ws, size_t ws_size,
                              hipStream_t stream) {
  (void)in_sizes; (void)n_in; (void)out_size; (void)ws_size;
  const float* x   = (const float*)d_in[0];   // [B,C,H,W] f32
  const float* sig = (const float*)d_in[1];   // [B,H,W]   f32
  float* out = (float*)d_out;                 // [B,C,H,W] f32

  // Workspace: interleaved transposed intermediates [B,C,W,H,2] (256 MiB).
  float* h12 = (float*)d_ws;

  dim3 blk(64);
  dim3 grid1(H_, W_ / TS, B_);
  dog_pass1<<<grid1, blk, 0, stream>>>(x, sig, h12);
  dim3 grid2(W_, H_ / TS, B_);
  dog_pass2<<<grid2, blk, 0, stream>>>(h12, sig, out);
}